// ImageNormalization2D_84645215469936
// MI455X (gfx1250) — compile-verified
//
#include <hip/hip_runtime.h>
#include <stdint.h>

#ifndef USE_ASYNC_LDS
#define USE_ASYNC_LDS 1
#endif

// Problem geometry (fixed by the reference).
#define NN 32
#define HH 512
#define WW 512
#define CC 2
#define FS 61
#define RAD 30
#define EPS 1e-7f

// Horizontal-pass tile geometry.
#define NROWS 16          // image rows per workgroup == WMMA M dimension
#define PADL 32           // zero halo each side for banded-WMMA A reads
#define ROWSTR 577        // 512 + 64 pad + 1 (odd stride -> no LDS bank conflicts)

typedef float v2f __attribute__((ext_vector_type(2)));
typedef float v8f __attribute__((ext_vector_type(8)));

// D(16x16) += A(16x4,f32) x B(4x16,f32). Arg pattern per CDNA5_HIP doc
// (8 args for 16x16x{4,32} float variants).
__device__ __forceinline__ v8f wmma4(v2f a, v2f b, v8f c) {
  return __builtin_amdgcn_wmma_f32_16x16x4_f32(
      /*neg_a=*/false, a, /*neg_b=*/false, b,
      /*c_mod=*/(short)0, c, /*reuse_a=*/false, /*reuse_b=*/false);
}

// Banded Toeplitz entry: 1.0 iff source column j contributes to output w
// under SAME zero padding (|j-w|<=30, j in image).
static __device__ __forceinline__ float inband(int j, int w) {
  return (j >= 0 && j < WW && j >= w - RAD && j <= w + RAD) ? 1.0f : 0.0f;
}

// ---------------------------------------------------------------------------
// Kernel 1: vertical 61-tap box sum of x -> V. One thread per flat (w,c)
// column, running-window sums over a 128-row h chunk. Wave32 lanes cover 32
// consecutive (w,c) floats -> one 128B line per step.
// ---------------------------------------------------------------------------
__global__ void __launch_bounds__(256) vert_box_x_kernel(
    const float* __restrict__ x, float* __restrict__ V) {
  const int cc = blockIdx.x * 256 + threadIdx.x;   // 0..1023 == 2*w + c
  const int n  = blockIdx.z;
  const int h1 = blockIdx.y * 128, h2 = h1 + 128;
  const size_t base = (size_t)n * HH * (WW * CC) + cc;

  float s = 0.0f;
  int lo = h1 - RAD; if (lo < 0) lo = 0;
  int hi = h1 + RAD; if (hi > HH - 1) hi = HH - 1;
  for (int j = lo; j <= hi; ++j) s += x[base + (size_t)j * (WW * CC)];

  for (int h = h1; h < h2; ++h) {
    V[base + (size_t)h * (WW * CC)] = s;
    const int add = h + RAD + 1;
    if (add < HH) s += x[base + (size_t)add * (WW * CC)];
    const int sub = h - RAD;
    if (sub >= 0) s -= x[base + (size_t)sub * (WW * CC)];
  }
}

// ---------------------------------------------------------------------------
// Kernel 2: per 16-row block: horizontal band-sum of V via WMMA -> full 2D
// box(x) -> d = x - box(x)*scale (written back over V: safe, row-for-row
// aliasing within the owning workgroup) -> horizontal band-sums of d and d^2
// via WMMA -> (Hd, Hd2) pairs.
// ---------------------------------------------------------------------------
__global__ void __launch_bounds__(256) horiz_wmma_kernel(
    const float* __restrict__ x, float* __restrict__ Vd,
    float2* __restrict__ hd) {
  extern __shared__ float smem[];
  float* VL = smem;                       // [CC][NROWS][ROWSTR] vertical sums
  float* DL = smem + CC * NROWS * ROWSTR; // [CC][NROWS][ROWSTR] d values

  const int tid = threadIdx.x;
  const int n  = blockIdx.y;
  const int h0 = blockIdx.x * NROWS;
  const float scale = 1.0f / (float)(FS * FS * CC);
  const size_t gb = ((size_t)n * HH + h0) * (WW * CC);

  // Zero the +-32 halos so banded-WMMA A reads are safe at image edges.
  for (int i = tid; i < CC * NROWS * 64; i += 256) {
    const int p  = i & 63;
    const int rc = i >> 6;
    const int wp = (p < PADL) ? p : (p + WW);
    VL[rc * ROWSTR + wp] = 0.0f;
    DL[rc * ROWSTR + wp] = 0.0f;
  }

  // Stage 16 rows of V into LDS, channel-deinterleaved.
#if USE_ASYNC_LDS
  {
    const uint64_t gbase = (uint64_t)(uintptr_t)(Vd + gb);
    for (int i = tid; i < NROWS * WW; i += 256) {
      const int r = i >> 9, w = i & (WW - 1);
      const unsigned voff = (unsigned)((r * (WW * CC) + 2 * w) * 4);
      const unsigned l0 =
          (unsigned)(uintptr_t)&VL[(0 * NROWS + r) * ROWSTR + PADL + w];
      const unsigned l1 =
          (unsigned)(uintptr_t)&VL[(1 * NROWS + r) * ROWSTR + PADL + w];
      asm volatile(
          "global_load_async_to_lds_b32 %0, %1, %2\n\t"
          "global_load_async_to_lds_b32 %3, %1, %2 offset:4"
          :: "v"(l0), "v"(voff), "s"(gbase), "v"(l1) : "memory");
    }
    asm volatile("s_wait_asynccnt 0" ::: "memory");
  }
#else
  for (int i = tid; i < NROWS * WW; i += 256) {
    const int r = i >> 9, w = i & (WW - 1);
    const float2 v = *(const float2*)(Vd + gb + (size_t)r * (WW * CC) + 2 * w);
    VL[(0 * NROWS + r) * ROWSTR + PADL + w] = v.x;
    VL[(1 * NROWS + r) * ROWSTR + PADL + w] = v.y;
  }
#endif
  __syncthreads();

  const int lane = tid & 31;
  const int wv   = tid >> 5;          // wave id 0..7
  const int ml   = lane & 15;         // M row (A) == N column (B/D)
  const int kg   = (lane >> 4) * 2;   // K sub-group: lanes<16 -> k{0,1}, else k{2,3}

  // Phase 1: Sxy = band(VL) -> d.  64 output tiles (2 ch x 32 w-tiles), 8/wave.
  for (int t = wv; t < CC * (WW / 16); t += 8) {
    const int c  = t >> 5;
    const int w0 = (t & 31) * 16;
    const float* Ar = &VL[(c * NROWS + ml) * ROWSTR + PADL];
    const int wn = w0 + ml;
    v8f acc = {0, 0, 0, 0, 0, 0, 0, 0};
    for (int ks = 0; ks < 20; ++ks) {   // K span [w0-32, w0+48): covers band +-30
      const int k0 = w0 - 32 + 4 * ks;
      v2f a; a.x = Ar[k0 + kg];            a.y = Ar[k0 + kg + 1];
      v2f b; b.x = inband(k0 + kg, wn);    b.y = inband(k0 + kg + 1, wn);
      acc = wmma4(a, b, acc);
    }
#pragma unroll
    for (int p = 0; p < 8; ++p) {       // D layout: VGPR p, lanes>=16 -> M+8
      const int m = p + ((lane < 16) ? 0 : 8);
      const int w = w0 + ml;
      const size_t gi = gb + (size_t)m * (WW * CC) + 2 * w + c;
      const float dv = x[gi] - acc[p] * scale;
      Vd[gi] = dv;                                       // d overwrites V
      DL[(c * NROWS + m) * ROWSTR + PADL + w] = dv;
    }
  }
  __syncthreads();

  // Phase 2: horizontal band-sums of d and d^2 -> (Hd, Hd2) pairs.
  for (int t = wv; t < CC * (WW / 16); t += 8) {
    const int c  = t >> 5;
    const int w0 = (t & 31) * 16;
    const float* Ar = &DL[(c * NROWS + ml) * ROWSTR + PADL];
    const int wn = w0 + ml;
    v8f s1 = {0, 0, 0, 0, 0, 0, 0, 0};
    v8f s2 = {0, 0, 0, 0, 0, 0, 0, 0};
    for (int ks = 0; ks < 20; ++ks) {
      const int k0 = w0 - 32 + 4 * ks;
      const float a0 = Ar[k0 + kg], a1 = Ar[k0 + kg + 1];
      v2f a;  a.x  = a0;       a.y  = a1;
      v2f aq; aq.x = a0 * a0;  aq.y = a1 * a1;
      v2f b;  b.x = inband(k0 + kg, wn);  b.y = inband(k0 + kg + 1, wn);
      s1 = wmma4(a,  b, s1);
      s2 = wmma4(aq, b, s2);
    }
#pragma unroll
    for (int p = 0; p < 8; ++p) {
      const int m = p + ((lane < 16) ? 0 : 8);
      const int w = w0 + ml;
      const size_t gi = gb + (size_t)m * (WW * CC) + 2 * w + c;
      hd[gi] = make_float2(s1[p], s2[p]);
    }
  }
}

// ---------------------------------------------------------------------------
// Kernel 3: vertical running-window sums of (Hd, Hd2) -> c1, c2; final
// out = d / (sqrt(max(c2 - c1^2, 0)) + eps).  float2 loads: 256B per wave.
// ---------------------------------------------------------------------------
__global__ void __launch_bounds__(256) vert_final_kernel(
    const float* __restrict__ dbuf, const float2* __restrict__ hd,
    float* __restrict__ out) {
  const int cc = blockIdx.x * 256 + threadIdx.x;
  const int n  = blockIdx.z;
  const int h1 = blockIdx.y * 128, h2 = h1 + 128;
  const size_t base = (size_t)n * HH * (WW * CC) + cc;
  const float scale = 1.0f / (float)(FS * FS * CC);

  float s1 = 0.0f, s2 = 0.0f;
  int lo = h1 - RAD; if (lo < 0) lo = 0;
  int hi = h1 + RAD; if (hi > HH - 1) hi = HH - 1;
  for (int j = lo; j <= hi; ++j) {
    const float2 v = hd[base + (size_t)j * (WW * CC)];
    s1 += v.x; s2 += v.y;
  }

  for (int h = h1; h < h2; ++h) {
    const float c1 = s1 * scale, c2 = s2 * scale;
    float var = c2 - c1 * c1;
    var = var > 0.0f ? var : 0.0f;
    const float stdv = sqrtf(var) + EPS;
    const size_t gi = base + (size_t)h * (WW * CC);
    out[gi] = dbuf[gi] / stdv;
    const int add = h + RAD + 1;
    if (add < HH) { const float2 v = hd[base + (size_t)add * (WW * CC)]; s1 += v.x; s2 += v.y; }
    const int sub = h - RAD;
    if (sub >= 0) { const float2 v = hd[base + (size_t)sub * (WW * CC)]; s1 -= v.x; s2 -= v.y; }
  }
}

extern "C" void kernel_launch(void* const* d_in, const int* in_sizes, int n_in,
                              void* d_out, int out_size, void* d_ws, size_t ws_size,
                              hipStream_t stream) {
  const float* x = (const float*)d_in[0];
  float* out = (float*)d_out;
  const size_t elems = (size_t)NN * HH * WW * CC;   // 16.7M floats

  // ws layout: [V then d, aliased: 64MB][Hd/Hd2 pairs: 128MB] = 192MB.
  if (ws_size < elems * sizeof(float) * 3) return;
  float* Vd = (float*)d_ws;
  float2* hd = (float2*)(Vd + elems);

  const size_t ldsBytes = (size_t)2 * CC * NROWS * ROWSTR * sizeof(float); // 147712
  (void)hipFuncSetAttribute((const void*)&horiz_wmma_kernel,
                            hipFuncAttributeMaxDynamicSharedMemorySize,
                            (int)ldsBytes);

  const dim3 blk(256, 1, 1);
  vert_box_x_kernel<<<dim3(4, 4, NN), blk, 0, stream>>>(x, Vd);
  horiz_wmma_kernel<<<dim3(HH / NROWS, NN), blk, ldsBytes, stream>>>(x, Vd, hd);
  vert_final_kernel<<<dim3(4, 4, NN), blk, 0, stream>>>(Vd, hd, out);
}